// SelfAttentionRPEConcat_32856499815188
// MI455X (gfx1250) — compile-verified
//
#include <hip/hip_runtime.h>
#include <math.h>

typedef __attribute__((ext_vector_type(2))) float v2f;
typedef __attribute__((ext_vector_type(8))) float v8f;

constexpr int kN     = 512;   // tokens
constexpr int kDim   = 256;   // model dim
constexpr int kHeads = 8;
constexpr int kDh    = 64;    // head dim
constexpr int kInner = 512;   // heads * head dim
constexpr int kRpe   = 66;    // (2*16+1)*2
#define PI_F 3.14159265358979323846f

// ---------------------------------------------------------------------------
// fp32 WMMA GEMM, 16x64 C-strip per wave: 4 accumulators share each A
// fragment (A reuse 4x), B fetches are coalesced 64-float row segments.
// K stepped by 4 via V_WMMA_F32_16X16X4_F32. N must be a multiple of 64.
// ---------------------------------------------------------------------------
__global__ __launch_bounds__(32)
void gemm16x64_f32(const float* __restrict__ A, const float* __restrict__ B,
                   const float* __restrict__ bias, float* __restrict__ C,
                   int K, int lda, int ldb, int ldc) {
  const int lo = threadIdx.x & 15;
  const int hi = threadIdx.x >> 4;
  const int n0 = blockIdx.x * 64;
  const int m0 = blockIdx.y * 16;

  const float* Arow = A + (m0 + lo) * lda + 2 * hi;   // A[m][kc + 2*hi + {0,1}]
  const float* Brow = B + (2 * hi) * ldb + n0 + lo;   // B[kc + 2*hi + {0,1}][n]

  v8f c0 = {}, c1 = {}, c2 = {}, c3 = {};
#pragma unroll 2
  for (int kc = 0; kc < K; kc += 4) {
    v2f a = *(const v2f*)(Arow + kc);
    const float* bp = Brow + kc * ldb;
    __builtin_prefetch(bp + 8 * ldb, 0, 1);           // global_prefetch_b8
    v2f b0, b1, b2, b3;
    b0.x = bp[0];  b0.y = bp[ldb + 0];
    b1.x = bp[16]; b1.y = bp[ldb + 16];
    b2.x = bp[32]; b2.y = bp[ldb + 32];
    b3.x = bp[48]; b3.y = bp[ldb + 48];
    c0 = __builtin_amdgcn_wmma_f32_16x16x4_f32(false, a, false, b0, (short)0, c0, false, false);
    c1 = __builtin_amdgcn_wmma_f32_16x16x4_f32(false, a, false, b1, (short)0, c1, false, false);
    c2 = __builtin_amdgcn_wmma_f32_16x16x4_f32(false, a, false, b2, (short)0, c2, false, false);
    c3 = __builtin_amdgcn_wmma_f32_16x16x4_f32(false, a, false, b3, (short)0, c3, false, false);
  }

  float bb0 = bias ? bias[n0 + lo +  0] : 0.0f;
  float bb1 = bias ? bias[n0 + lo + 16] : 0.0f;
  float bb2 = bias ? bias[n0 + lo + 32] : 0.0f;
  float bb3 = bias ? bias[n0 + lo + 48] : 0.0f;
  float* Cr = C + n0 + lo;
#pragma unroll
  for (int r = 0; r < 8; ++r) {
    float* crow = Cr + (m0 + hi * 8 + r) * ldc;
    crow[0]  = c0[r] + bb0;
    crow[16] = c1[r] + bb1;
    crow[32] = c2[r] + bb2;
    crow[48] = c3[r] + bb3;
  }
}

// ---------------------------------------------------------------------------
// t[i,h,r] = sum_e q[i, h*64+e] * Wk[256 + r, h*64 + e]
// ---------------------------------------------------------------------------
__global__ void t_proj(const float* __restrict__ q, const float* __restrict__ Wk,
                       float* __restrict__ t) {
  int idx = blockIdx.x * blockDim.x + threadIdx.x;
  if (idx >= kN * kHeads * kRpe) return;
  int r = idx % kRpe;
  int h = (idx / kRpe) % kHeads;
  int i = idx / (kRpe * kHeads);
  const float* qp = q + i * kInner + h * kDh;
  const float* wp = Wk + (kDim + r) * kInner + h * kDh;
  float s = 0.0f;
#pragma unroll 8
  for (int e = 0; e < kDh; ++e) s = fmaf(qp[e], wp[e], s);
  t[idx] = s;
}

// ---------------------------------------------------------------------------
// RPE dot for one axis: features are [sin(th_b)]_{b<16}, [cos(th_b)]_{b<16}, d
// with th_b = d*pi*(1 + 7b/15) — equally spaced angles -> rotation recurrence:
// 4 transcendentals instead of 32.
// ---------------------------------------------------------------------------
__device__ __forceinline__ float rpe_axis_dot(float d, const float* __restrict__ T) {
  float sum = d * T[32];
  float th0 = d * PI_F;
  float dl  = th0 * (7.0f / 15.0f);
  float s = __sinf(th0), c = __cosf(th0);
  float sd = __sinf(dl), cd = __cosf(dl);
#pragma unroll
  for (int b = 0; b < 16; ++b) {
    sum = fmaf(s, T[b], sum);
    sum = fmaf(c, T[16 + b], sum);
    float s2 = fmaf(s, cd,  c * sd);
    float c2 = fmaf(c, cd, -s * sd);
    s = s2; c = c2;
  }
  return sum;
}

// ---------------------------------------------------------------------------
// Fused flash attention with RPE score correction.
// One wave per (head, 16-query tile); loops over 32 key tiles.
// ---------------------------------------------------------------------------
__global__ __launch_bounds__(32)
void attn_rpe(const float* __restrict__ q, const float* __restrict__ k,
              const float* __restrict__ v, const float* __restrict__ t,
              const float* __restrict__ pos, float* __restrict__ out) {
  __shared__ float ldsT[16 * kRpe];   // per-row 66-dim RPE projection of q
  __shared__ float ldsP[16 * 16];     // softmax probabilities (re-fragment for PV)
  __shared__ float ldsPi[16 * 2];     // query positions

  const int h    = blockIdx.x >> 5;
  const int i0   = (blockIdx.x & 31) * 16;
  const int lane = threadIdx.x;
  const int lo   = lane & 15;
  const int hi   = lane >> 4;

  for (int idx = lane; idx < 16 * kRpe; idx += 32) {
    int row = idx / kRpe;
    int r   = idx % kRpe;
    ldsT[idx] = t[((i0 + row) * kHeads + h) * kRpe + r];
  }
  if (lane < 16) {
    ldsPi[lane * 2 + 0] = pos[(i0 + lane) * 2 + 0];
    ldsPi[lane * 2 + 1] = pos[(i0 + lane) * 2 + 1];
  }
  __syncthreads();

  // Preload Q tile fragments: 16 K-chunks of 4.
  v2f a[16];
  const float* Qrow = q + (i0 + lo) * kInner + h * kDh + 2 * hi;
#pragma unroll
  for (int kc = 0; kc < 16; ++kc) a[kc] = *(const v2f*)(Qrow + kc * 4);

  float m[8], l[8];
#pragma unroll
  for (int r = 0; r < 8; ++r) { m[r] = -INFINITY; l[r] = 0.0f; }
  v8f acc0 = {}, acc1 = {}, acc2 = {}, acc3 = {};

  for (int jt = 0; jt < 32; ++jt) {
    const int j0 = jt * 16;

    // ---- S tile = Q @ K^T (16 WMMAs over d = 64) ----
    v8f c = {};
    const float* Krow = k + (j0 + lo) * kInner + h * kDh + 2 * hi;
#pragma unroll
    for (int kc = 0; kc < 16; ++kc) {
      v2f b = *(const v2f*)(Krow + kc * 4);
      c = __builtin_amdgcn_wmma_f32_16x16x4_f32(false, a[kc], false, b, (short)0,
                                                c, false, false);
    }

    // ---- RPE correction + online softmax ----
    float pjx = pos[(j0 + lo) * 2 + 0];
    float pjy = pos[(j0 + lo) * 2 + 1];
    float alpha[8];
#pragma unroll
    for (int r = 0; r < 8; ++r) {
      const int row = hi * 8 + r;
      float dx = (pjx - ldsPi[row * 2 + 0]) * (1.0f / 51.0f);
      float dy = (pjy - ldsPi[row * 2 + 1]) * (1.0f / 51.0f);
      dx = dx / (1.0f + fabsf(dx));
      dy = dy / (1.0f + fabsf(dy));
      const float* Trow = &ldsT[row * kRpe];
      float rp = rpe_axis_dot(dx, Trow) + rpe_axis_dot(dy, Trow + 33);
      float sv = (c[r] + rp) * 0.125f;  // scale = 64^-0.5

      float rmax = sv;                   // reduce over 16 key columns (lane half)
      rmax = fmaxf(rmax, __shfl_xor(rmax, 1, 16));
      rmax = fmaxf(rmax, __shfl_xor(rmax, 2, 16));
      rmax = fmaxf(rmax, __shfl_xor(rmax, 4, 16));
      rmax = fmaxf(rmax, __shfl_xor(rmax, 8, 16));
      float mnew = fmaxf(m[r], rmax);
      float p = __expf(sv - mnew);
      float rsum = p;
      rsum += __shfl_xor(rsum, 1, 16);
      rsum += __shfl_xor(rsum, 2, 16);
      rsum += __shfl_xor(rsum, 4, 16);
      rsum += __shfl_xor(rsum, 8, 16);
      float al = __expf(m[r] - mnew);
      l[r] = l[r] * al + rsum;
      m[r] = mnew;
      alpha[r] = al;
      ldsP[row * 16 + lo] = p;
    }
#pragma unroll
    for (int r = 0; r < 8; ++r) {
      acc0[r] *= alpha[r]; acc1[r] *= alpha[r];
      acc2[r] *= alpha[r]; acc3[r] *= alpha[r];
    }
    __syncthreads();

    // ---- O += P @ V (A frag from LDS, B frag from V rows; 16 WMMAs) ----
#pragma unroll
    for (int kc = 0; kc < 4; ++kc) {
      v2f pa = *(const v2f*)&ldsP[lo * 16 + kc * 4 + 2 * hi];
      const float* Vr0 = v + (j0 + kc * 4 + 2 * hi) * kInner + h * kDh + lo;
      const float* Vr1 = Vr0 + kInner;
      v2f b0; b0.x = Vr0[0];  b0.y = Vr1[0];
      v2f b1; b1.x = Vr0[16]; b1.y = Vr1[16];
      v2f b2; b2.x = Vr0[32]; b2.y = Vr1[32];
      v2f b3; b3.x = Vr0[48]; b3.y = Vr1[48];
      acc0 = __builtin_amdgcn_wmma_f32_16x16x4_f32(false, pa, false, b0, (short)0, acc0, false, false);
      acc1 = __builtin_amdgcn_wmma_f32_16x16x4_f32(false, pa, false, b1, (short)0, acc1, false, false);
      acc2 = __builtin_amdgcn_wmma_f32_16x16x4_f32(false, pa, false, b2, (short)0, acc2, false, false);
      acc3 = __builtin_amdgcn_wmma_f32_16x16x4_f32(false, pa, false, b3, (short)0, acc3, false, false);
    }
    __syncthreads();
  }

  float inv[8];
#pragma unroll
  for (int r = 0; r < 8; ++r) inv[r] = 1.0f / l[r];
  float* Ocol = out + h * kDh + lo;
#pragma unroll
  for (int r = 0; r < 8; ++r) {
    int grow = i0 + hi * 8 + r;
    Ocol[grow * kInner + 0]  = acc0[r] * inv[r];
    Ocol[grow * kInner + 16] = acc1[r] * inv[r];
    Ocol[grow * kInner + 32] = acc2[r] * inv[r];
    Ocol[grow * kInner + 48] = acc3[r] * inv[r];
  }
}

// ---------------------------------------------------------------------------
extern "C" void kernel_launch(void* const* d_in, const int* in_sizes, int n_in,
                              void* d_out, int out_size, void* d_ws, size_t ws_size,
                              hipStream_t stream) {
  const float* x   = (const float*)d_in[0];
  const float* pos = (const float*)d_in[1];
  const float* Wq  = (const float*)d_in[2];
  const float* Wk  = (const float*)d_in[3];  // (256+66) x 512
  const float* Wv  = (const float*)d_in[4];
  const float* Wo  = (const float*)d_in[5];
  const float* bo  = (const float*)d_in[6];
  float* out = (float*)d_out;

  float* ws   = (float*)d_ws;
  float* q    = ws;                          // 512*512
  float* kbuf = q    + kN * kInner;          // 512*512
  float* vbuf = kbuf + kN * kInner;          // 512*512
  float* tbuf = vbuf + kN * kInner;          // 512*8*66
  float* attn = tbuf + kN * kHeads * kRpe;   // 512*512

  dim3 wave(32);

  // Q, K(content), V projections: (512x256) @ (256x512)
  dim3 gProj(kInner / 64, kN / 16);
  gemm16x64_f32<<<gProj, wave, 0, stream>>>(x, Wq, (const float*)nullptr, q,
                                            kDim, kDim, kInner, kInner);
  gemm16x64_f32<<<gProj, wave, 0, stream>>>(x, Wk, (const float*)nullptr, kbuf,
                                            kDim, kDim, kInner, kInner);
  gemm16x64_f32<<<gProj, wave, 0, stream>>>(x, Wv, (const float*)nullptr, vbuf,
                                            kDim, kDim, kInner, kInner);

  // t[i,h,r] = q_ih . Wk_r[r, h-slice]
  int tElems = kN * kHeads * kRpe;
  t_proj<<<(tElems + 255) / 256, 256, 0, stream>>>(q, Wk, tbuf);

  // Fused flash attention with RPE scores.
  attn_rpe<<<dim3(kHeads * (kN / 16)), wave, 0, stream>>>(q, kbuf, vbuf, tbuf,
                                                          pos, attn);

  // Output projection: (512x512) @ (512x256) + bias
  dim3 gOut(kDim / 64, kN / 16);
  gemm16x64_f32<<<gOut, wave, 0, stream>>>(attn, Wo, bo, out,
                                           kInner, kInner, kDim, kDim);
}